// SensorGNN_75110388072633
// MI455X (gfx1250) — compile-verified
//
#include <hip/hip_runtime.h>
#include <math.h>

typedef __attribute__((ext_vector_type(2))) float v2f;
typedef __attribute__((ext_vector_type(8))) float v8f;

#define HID 64

// ---------------- init / utility kernels ----------------

__global__ void gcn_init_kernel(float* deg, float* agg, float* acc6,
                                int n, long nagg) {
  long stride = (long)gridDim.x * blockDim.x;
  long tid = (long)blockIdx.x * blockDim.x + threadIdx.x;
  for (long i = tid; i < nagg; i += stride) agg[i] = 0.0f;
  for (long i = tid; i < n; i += stride) deg[i] = 1.0f;  // self-loop contributes 1
  if (tid < 8) acc6[tid] = 0.0f;
}

__global__ void gcn_zero_kernel(float* p, long n) {
  long stride = (long)gridDim.x * blockDim.x;
  for (long i = (long)blockIdx.x * blockDim.x + threadIdx.x; i < n; i += stride)
    p[i] = 0.0f;
}

__global__ void gcn_degree_kernel(const int* __restrict__ dst, float* deg, int E) {
  int e = blockIdx.x * blockDim.x + threadIdx.x;
  if (e < E) atomicAdd(&deg[dst[e]], 1.0f);
}

__global__ void gcn_dinv_kernel(float* deg, int n) {
  int i = blockIdx.x * blockDim.x + threadIdx.x;
  if (i < n) deg[i] = rsqrtf(deg[i]);  // deg >= 1 always (self-loops)
}

// ---------------- WMMA GEMM: H[M,64] = X[M,K] * W[K,64] ----------------
// One wave per 16-row tile; 4 accumulators cover N=0..63.
// fp32 WMMA (16x16x4): A lanes 0-15 -> K={k0,k0+1}, lanes 16-31 -> K={k0+2,k0+3}.

__global__ void gcn_gemm_wmma_kernel(const float* __restrict__ X,
                                     const float* __restrict__ W,
                                     float* __restrict__ H,
                                     int M, int K) {
  const int lane = threadIdx.x & 31;
  const int wave = threadIdx.x >> 5;
  const int rowTile = blockIdx.x * (blockDim.x >> 5) + wave;
  if (rowTile * 16 >= M) return;  // wave-uniform exit: EXEC stays all-ones for WMMA

  const int mrow = rowTile * 16 + (lane & 15);
  const long m = (mrow < M) ? mrow : (M - 1);  // clamp keeps EXEC full
  const int khalf = (lane >> 4) << 1;          // 0 or 2
  const int ncol = lane & 15;

  v8f acc0 = {0.f,0.f,0.f,0.f,0.f,0.f,0.f,0.f};
  v8f acc1 = {0.f,0.f,0.f,0.f,0.f,0.f,0.f,0.f};
  v8f acc2 = {0.f,0.f,0.f,0.f,0.f,0.f,0.f,0.f};
  v8f acc3 = {0.f,0.f,0.f,0.f,0.f,0.f,0.f,0.f};

  for (int k0 = 0; k0 < K; k0 += 4) {
    const float* ap = X + m * (long)K + k0 + khalf;
    v2f a;
    a.x = ap[0];
    a.y = ap[1];

    const float* wp = W + (long)(k0 + khalf) * HID + ncol;
    v2f b;

    b.x = wp[0];        b.y = wp[HID];
    acc0 = __builtin_amdgcn_wmma_f32_16x16x4_f32(false, a, false, b, (short)0,
                                                 acc0, false, false);
    b.x = wp[16];       b.y = wp[HID + 16];
    acc1 = __builtin_amdgcn_wmma_f32_16x16x4_f32(false, a, false, b, (short)0,
                                                 acc1, false, false);
    b.x = wp[32];       b.y = wp[HID + 32];
    acc2 = __builtin_amdgcn_wmma_f32_16x16x4_f32(false, a, false, b, (short)0,
                                                 acc2, false, false);
    b.x = wp[48];       b.y = wp[HID + 48];
    acc3 = __builtin_amdgcn_wmma_f32_16x16x4_f32(false, a, false, b, (short)0,
                                                 acc3, false, false);
  }

  // C/D layout: VGPR r -> rows (r, r+8) split across lane halves; col = lane&15.
  const int rbase = (lane >> 4) << 3;  // 0 or 8
#pragma unroll
  for (int r = 0; r < 8; ++r) {
    int mm = rowTile * 16 + rbase + r;
    if (mm < M) {
      float* op = H + (long)mm * HID + ncol;
      op[0]  = acc0[r];
      op[16] = acc1[r];
      op[32] = acc2[r];
      op[48] = acc3[r];
    }
  }
}

// ---------------- edge scatter: agg[dst] += h[src] * dinv[src]*dinv[dst] ----------------

__global__ void gcn_scatter_kernel(const float* __restrict__ H,
                                   const int* __restrict__ src,
                                   const int* __restrict__ dst,
                                   const float* __restrict__ dinv,
                                   float* agg, int E) {
  int e = blockIdx.x * (blockDim.x >> 5) + (threadIdx.x >> 5);
  if (e >= E) return;
  int lane = threadIdx.x & 31;
  int s = src[e];
  int d = dst[e];
  float nrm = dinv[s] * dinv[d];
  long sb = (long)s * HID;
  long db = (long)d * HID;
  atomicAdd(&agg[db + lane],      H[sb + lane] * nrm);
  atomicAdd(&agg[db + lane + 32], H[sb + lane + 32] * nrm);
}

// ---------------- self-loop + bias + relu (in place on agg) ----------------

__global__ void gcn_self_bias_relu_kernel(float* agg,
                                          const float* __restrict__ H,
                                          const float* __restrict__ dinv,
                                          const float* __restrict__ bias,
                                          long n) {
  long i = (long)blockIdx.x * blockDim.x + threadIdx.x;
  if (i >= n) return;
  int node = (int)(i >> 6);
  int f = (int)(i & 63);
  float di = dinv[node];
  float v = agg[i] + H[i] * di * di + bias[f];
  agg[i] = v > 0.0f ? v : 0.0f;
}

// ---------------- classifier: logits[c] = sum_i H.flat[i] * Wc[i][c] ----------------

__global__ void gcn_classifier_kernel(const float* __restrict__ H,
                                      const float* __restrict__ Wc,
                                      float* acc6, long total) {
  float a0 = 0.f, a1 = 0.f, a2 = 0.f, a3 = 0.f, a4 = 0.f, a5 = 0.f;
  long stride = (long)gridDim.x * blockDim.x;
  for (long i = (long)blockIdx.x * blockDim.x + threadIdx.x; i < total; i += stride) {
    float h = H[i];
    const float* w = Wc + i * 6;
    __builtin_prefetch(w + stride * 6, 0, 1);  // global_prefetch_b8 on the Wc stream
    a0 += h * w[0];
    a1 += h * w[1];
    a2 += h * w[2];
    a3 += h * w[3];
    a4 += h * w[4];
    a5 += h * w[5];
  }
#pragma unroll
  for (int off = 16; off > 0; off >>= 1) {  // wave32 reduction
    a0 += __shfl_down(a0, off);
    a1 += __shfl_down(a1, off);
    a2 += __shfl_down(a2, off);
    a3 += __shfl_down(a3, off);
    a4 += __shfl_down(a4, off);
    a5 += __shfl_down(a5, off);
  }
  if ((threadIdx.x & 31) == 0) {
    atomicAdd(&acc6[0], a0);
    atomicAdd(&acc6[1], a1);
    atomicAdd(&acc6[2], a2);
    atomicAdd(&acc6[3], a3);
    atomicAdd(&acc6[4], a4);
    atomicAdd(&acc6[5], a5);
  }
}

__global__ void gcn_softmax_kernel(const float* __restrict__ acc6,
                                   const float* __restrict__ bc,
                                   float* out) {
  if (threadIdx.x == 0) {
    float l[6];
    float mx = -3.4e38f;
#pragma unroll
    for (int c = 0; c < 6; ++c) {
      l[c] = acc6[c] + bc[c];
      mx = l[c] > mx ? l[c] : mx;
    }
    float s = 0.f;
#pragma unroll
    for (int c = 0; c < 6; ++c) {
      l[c] = expf(l[c] - mx);
      s += l[c];
    }
    float inv = 1.0f / s;
#pragma unroll
    for (int c = 0; c < 6; ++c) out[c] = l[c] * inv;
  }
}

// ---------------- launcher ----------------

extern "C" void kernel_launch(void* const* d_in, const int* in_sizes, int n_in,
                              void* d_out, int out_size, void* d_ws, size_t ws_size,
                              hipStream_t stream) {
  const float* x  = (const float*)d_in[0];
  const int*   ei = (const int*)d_in[1];
  const float* W1 = (const float*)d_in[2];
  const float* b1 = (const float*)d_in[3];
  const float* W2 = (const float*)d_in[4];
  const float* b2 = (const float*)d_in[5];
  const float* Wc = (const float*)d_in[6];
  const float* bc = (const float*)d_in[7];
  float* out = (float*)d_out;

  const int F = 128;
  const int N = in_sizes[0] / F;   // 50000
  const int E = in_sizes[1] / 2;   // 800000
  const int* src = ei;
  const int* dst = ei + E;
  const long nh = (long)N * HID;   // 3.2M

  float* dinv = (float*)d_ws;      // [N]
  float* hbuf = dinv + N;          // [N*64]
  float* agg  = hbuf + nh;         // [N*64]
  float* acc6 = agg + nh;          // [8]

  // normalization
  gcn_init_kernel<<<2048, 256, 0, stream>>>(dinv, agg, acc6, N, nh);
  gcn_degree_kernel<<<(E + 255) / 256, 256, 0, stream>>>(dst, dinv, E);
  gcn_dinv_kernel<<<(N + 255) / 256, 256, 0, stream>>>(dinv, N);

  const int rowTiles = (N + 15) / 16;              // 3125
  const int gemmBlocks = (rowTiles + 3) / 4;       // 4 waves/block

  // layer 1
  gcn_gemm_wmma_kernel<<<gemmBlocks, 128, 0, stream>>>(x, W1, hbuf, N, F);
  gcn_scatter_kernel<<<(E + 7) / 8, 256, 0, stream>>>(hbuf, src, dst, dinv, agg, E);
  gcn_self_bias_relu_kernel<<<(int)((nh + 255) / 256), 256, 0, stream>>>(agg, hbuf, dinv, b1, nh);

  // layer 2 (input = agg, output gemm -> hbuf, re-aggregate into agg)
  gcn_gemm_wmma_kernel<<<gemmBlocks, 128, 0, stream>>>(agg, W2, hbuf, N, HID);
  gcn_zero_kernel<<<2048, 256, 0, stream>>>(agg, nh);
  gcn_scatter_kernel<<<(E + 7) / 8, 256, 0, stream>>>(hbuf, src, dst, dinv, agg, E);
  gcn_self_bias_relu_kernel<<<(int)((nh + 255) / 256), 256, 0, stream>>>(agg, hbuf, dinv, b2, nh);

  // classifier + softmax
  gcn_classifier_kernel<<<4096, 256, 0, stream>>>(agg, Wc, acc6, nh);
  gcn_softmax_kernel<<<1, 32, 0, stream>>>(acc6, bc, out);
}